// InfoNCESimilarityLoss_56066503082769
// MI455X (gfx1250) — compile-verified
//
#include <hip/hip_runtime.h>
#include <hip/hip_bf16.h>
#include <math.h>

// ---------------------------------------------------------------------------
// InfoNCE patch-similarity loss for MI455X (gfx1250, wave32, WMMA).
//
// Shapes: B=16, C=256, H=W=224, p=14 -> 16x16=256 patches, D=C*p*p=50176.
// Stage 1 (memory-bound streaming, ~822MB of ref_feature):
//   grid (CSPLIT=32, nH=16, B=16), block = 1 wave (32 lanes).
//   Each wave: 16 patches (one patch-grid row) x 8 channels, accumulating
//     dot(t_patch, q_patch) via V_WMMA_F32_16X16X4_F32 (A = 16x4 ref tile,
//     B = query chunk replicated over all 16 columns),
//     ||t||^2 and ||q||^2 via VALU, folded across lane halves with shfl_xor.
//   The 14-wide patch row = 3 unconditional K=4 chunks + 1 remainder chunk
//   whose invalid half is killed with a hoisted multiplicative mask (no
//   per-iteration cndmask / address selects).
// Stage 2: single block reduces partials, cosine-sim, L2-normalize over the
//   256-patch grid, log-softmax, mean NLL -> d_out[0].
// ---------------------------------------------------------------------------

typedef float v2f __attribute__((ext_vector_type(2)));
typedef float v8f __attribute__((ext_vector_type(8)));

#define Bc 16
#define Cc 256
#define Hc 224
#define Wc 224
#define Pp 14
#define NHp 16
#define NWp 16
#define Np (NHp * NWp)
#define CSPLIT 32
#define CPW (Cc / CSPLIT)   // 8 channels per wave
#define EPS_COS 1e-8f
#define EPS_NORM 1e-12f

__global__ __launch_bounds__(32)
void infonce_stage1(const float* __restrict__ qf,   // query_feature [B,C,H,W]
                    const float* __restrict__ tf,   // ref_feature   [B,C,H,W]
                    const int*   __restrict__ click,// [B,2] (x,y)
                    float* __restrict__ wdot,       // [B][256][CSPLIT]
                    float* __restrict__ wnrm,       // [B][256][CSPLIT]
                    float* __restrict__ wq)         // [B][CSPLIT]
{
    const int split = blockIdx.x;        // channel slice
    const int ig    = blockIdx.y;        // patch-grid row
    const int b     = blockIdx.z;        // batch
    const int lane  = threadIdx.x;       // 0..31
    const int m     = lane & 15;         // patch column j / matrix row M
    const int half  = lane >> 4;         // 0: K=0,1  1: K=2,3 (A/B layout)
    const float hmask = (half == 0) ? 1.0f : 0.0f;  // remainder-chunk mask

    const int cx = click[b * 2 + 0];
    const int cy = click[b * 2 + 1];
    const int qh = cy / Pp;
    const int qw = cx / Pp;

    const size_t imgBase = (size_t)b * Cc * Hc * Wc;
    const float* tb = tf + imgBase + (size_t)(ig * Pp) * Wc + m * Pp;
    const float* qb = qf + imgBase + (size_t)(qh * Pp) * Wc + qw * Pp;

    v8f  acc  = {};        // 16x16 f32 accumulator (every column identical)
    float nacc = 0.0f;     // partial ||t_row||^2 for this lane's K slots
    float qacc = 0.0f;     // partial ||q||^2 (identical within lane half)

    const int c0 = split * CPW;
    for (int c = c0; c < c0 + CPW; ++c) {
        const float* tc = tb + (size_t)c * (Hc * Wc);
        const float* qc = qb + (size_t)c * (Hc * Wc);
        for (int dy = 0; dy < Pp; ++dy) {
            const float* tr = tc + dy * Wc;
            const float* qr = qc + dy * Wc;
            // stream hint: next image row of the ref stream
            __builtin_prefetch(tr + Wc, 0, 1);
            // 3 full K=4 chunks: elements 0..11 -- no masking needed
#pragma unroll
            for (int ch = 0; ch < 3; ++ch) {
                const int k0 = ch * 4 + 2 * half;       // <= 10, always valid
                v2f ta = *(const v2f*)(tr + k0);
                v2f qa = *(const v2f*)(qr + k0);
                acc = __builtin_amdgcn_wmma_f32_16x16x4_f32(
                        false, ta, false, qa, (short)0, acc, false, false);
                nacc += ta.x * ta.x + ta.y * ta.y;
                qacc += qa.x * qa.x + qa.y * qa.y;
            }
            // remainder chunk: elements 12,13 (half 1 slots are padding)
            {
                v2f ta = *(const v2f*)(tr + 12);        // both halves: safe addr
                v2f qa = *(const v2f*)(qr + 12);
                qa.x *= hmask;                          // B=0 on half 1 -> +0
                qa.y *= hmask;
                acc = __builtin_amdgcn_wmma_f32_16x16x4_f32(
                        false, ta, false, qa, (short)0, acc, false, false);
                nacc = fmaf(hmask, ta.x * ta.x + ta.y * ta.y, nacc);
                qacc += qa.x * qa.x + qa.y * qa.y;      // already masked
            }
        }
    }

    // fold lane-halves: row M's norm is split between lane m and lane m+16
    const float nsum = nacc + __shfl_xor(nacc, 16, 32);
    const float qsum = qacc + __shfl_xor(qacc, 16, 32);

    const size_t nBase = (size_t)b * Np + ig * NWp;
    // C/D layout: lane 0 vgprs 0..7 = rows 0..7; lane 16 = rows 8..15
    if (lane == 0 || lane == 16) {
        const int rowOff = (lane == 16) ? 8 : 0;
#pragma unroll
        for (int r = 0; r < 8; ++r)
            wdot[(nBase + rowOff + r) * CSPLIT + split] = acc[r];
    }
    if (lane < 16)
        wnrm[(nBase + m) * CSPLIT + split] = nsum;
    if (lane == 0 && ig == 0)
        wq[b * CSPLIT + split] = qsum;
}

__global__ __launch_bounds__(256)
void infonce_stage2(const float* __restrict__ wdot,
                    const float* __restrict__ wnrm,
                    const float* __restrict__ wq,
                    const float* __restrict__ scale_p,
                    const int*   __restrict__ gt,
                    float* __restrict__ out)
{
    __shared__ float red[Np];
    __shared__ float sh0;
    const int n = threadIdx.x;           // patch index 0..255
    const float scale = scale_p[0];
    float loss_sum = 0.0f;

    for (int b = 0; b < Bc; ++b) {
        float dot = 0.0f, nrm = 0.0f;
        const size_t base = ((size_t)b * Np + n) * CSPLIT;
#pragma unroll 4
        for (int s = 0; s < CSPLIT; ++s) { dot += wdot[base + s]; nrm += wnrm[base + s]; }
        if (n == 0) {
            float qs = 0.0f;
            for (int s = 0; s < CSPLIT; ++s) qs += wq[b * CSPLIT + s];
            sh0 = qs;
        }
        __syncthreads();
        const float qn  = fmaxf(sqrtf(sh0), EPS_COS);
        const float tn  = fmaxf(sqrtf(nrm), EPS_COS);
        float sim = dot / (tn * qn);
        __syncthreads();

        // L2-normalize sim over the 256-patch grid
        red[n] = sim * sim; __syncthreads();
        for (int o = 128; o > 0; o >>= 1) { if (n < o) red[n] += red[n + o]; __syncthreads(); }
        const float l2 = fmaxf(sqrtf(red[0]), EPS_NORM); __syncthreads();
        const float logit = scale * (sim / l2);

        // log-softmax: max then sum-exp
        red[n] = logit; __syncthreads();
        for (int o = 128; o > 0; o >>= 1) { if (n < o) red[n] = fmaxf(red[n], red[n + o]); __syncthreads(); }
        const float mx = red[0]; __syncthreads();
        red[n] = expf(logit - mx); __syncthreads();
        for (int o = 128; o > 0; o >>= 1) { if (n < o) red[n] += red[n + o]; __syncthreads(); }
        const float se = red[0]; __syncthreads();

        const int gx = gt[b * 2 + 0], gy = gt[b * 2 + 1];
        const int lab = (gy / Pp) * NWp + (gx / Pp);
        if (n == lab) sh0 = logit;
        __syncthreads();
        if (n == 0) loss_sum += -(sh0 - mx - logf(se));
        __syncthreads();
    }
    if (n == 0) out[0] = loss_sum / (float)Bc;
}

extern "C" void kernel_launch(void* const* d_in, const int* in_sizes, int n_in,
                              void* d_out, int out_size, void* d_ws, size_t ws_size,
                              hipStream_t stream) {
    const float* qf    = (const float*)d_in[0];
    const float* tf    = (const float*)d_in[1];
    const float* scale = (const float*)d_in[2];
    const int*   click = (const int*)d_in[3];
    const int*   gt    = (const int*)d_in[4];
    (void)in_sizes; (void)n_in; (void)out_size; (void)ws_size;

    float* wdot = (float*)d_ws;                         // 16*256*32 floats
    float* wnrm = wdot + (size_t)Bc * Np * CSPLIT;      // 16*256*32 floats
    float* wq   = wnrm + (size_t)Bc * Np * CSPLIT;      // 16*32 floats

    dim3 grid1(CSPLIT, NHp, Bc);
    infonce_stage1<<<grid1, 32, 0, stream>>>(qf, tf, click, wdot, wnrm, wq);
    infonce_stage2<<<1, Np, 0, stream>>>(wdot, wnrm, wq, scale, gt, (float*)d_out);
}